// CDD_34952443855327
// MI455X (gfx1250) — compile-verified
//
#include <hip/hip_runtime.h>

#define N_USER   50000
#define N_TOT    100000
#define DIN      128
#define NNZ_C    3200000
#define BATCH    4096
#define NEG_SLOPE 0.2f
#define ASTRIDE  132   // LDS row stride (floats): bank-conflict-free WMMA A reads

typedef __attribute__((ext_vector_type(2))) float v2f;
typedef __attribute__((ext_vector_type(8))) float v8f;

// ego0 = [[user_emb | feature] ; [item_emb1 | item_emb2]]  -> (N, 128)
__global__ void build_ego_kernel(const float* __restrict__ user_emb,
                                 const float* __restrict__ item_emb1,
                                 const float* __restrict__ item_emb2,
                                 const float* __restrict__ feature,
                                 float* __restrict__ ego) {
  long i4   = (long)blockIdx.x * blockDim.x + threadIdx.x;
  long elem = i4 * 4;
  long row  = elem >> 7;
  int  col  = (int)(elem & 127);
  const float* src;
  if (col < 64) {
    src = (row < N_USER) ? user_emb + row * 64 + col
                         : item_emb1 + (row - N_USER) * 64 + col;
  } else {
    int c2 = col - 64;
    src = (row < N_USER) ? feature + row * 64 + c2
                         : item_emb2 + (row - N_USER) * 64 + c2;
  }
  *(float4*)(ego + elem) = *(const float4*)src;
}

// comb = w0*v + w1*v^2 + w2*v^6
__global__ void comb_kernel(const float* __restrict__ vals,
                            const float* __restrict__ w,
                            float* __restrict__ comb) {
  int i = blockIdx.x * blockDim.x + threadIdx.x;
  float v  = vals[i];
  float v2 = v * v;
  comb[i] = w[0] * v + w[1] * v2 + w[2] * (v2 * v2 * v2);
}

// wave-per-edge SpMM scatter: Y[row] += comb[e] * X[col]; lane owns a float4 slice.
// Edge index made provably wave-uniform so metadata loads take the scalar path.
__global__ void spmm_kernel(const float* __restrict__ comb,
                            const int* __restrict__ rows,
                            const int* __restrict__ cols,
                            const float* __restrict__ X,
                            float* __restrict__ Y) {
  int e    = __builtin_amdgcn_readfirstlane(blockIdx.x * 8 + (threadIdx.x >> 5));
  int lane = threadIdx.x & 31;
  float v  = comb[e];
  long col = cols[e];
  long row = rows[e];
  float4 x = *(const float4*)(X + col * DIN + lane * 4);
  float* y = Y + row * DIN + lane * 4;
  atomicAdd(y + 0, v * x.x);
  atomicAdd(y + 1, v * x.y);
  atomicAdd(y + 2, v * x.z);
  atomicAdd(y + 3, v * x.w);
}

// Fused: leakyrelu(side@Wgc + (ego*side)@Wbi + bgc + bbi) -> out_raw (next-layer ego)
// and row-L2-normalized copy -> out_norm.
// Block = 32 output rows (two 16-row M-tiles); 8 waves cover the 128 columns.
// Each B fragment pair feeds TWO wmma ops (one per M-tile) -> halved B traffic.
__global__ void __launch_bounds__(256)
gemm_layer_kernel(const float* __restrict__ side, const float* __restrict__ ego,
                  const float* __restrict__ Wgc,  const float* __restrict__ bgc,
                  const float* __restrict__ Wbi,  const float* __restrict__ bbi,
                  float* __restrict__ out_raw,    float* __restrict__ out_norm) {
  __shared__ float A1[32 * ASTRIDE];   // side tiles
  __shared__ float A2[32 * ASTRIDE];   // ego*side tiles
  __shared__ float rowss[32];          // per-row sum of squares

  const int  tid  = threadIdx.x;
  const long base = (long)blockIdx.x * (32 * DIN);

  for (int i = tid * 4; i < 32 * DIN; i += 1024) {
    int r = i >> 7, c = i & 127;
    float4 s = *(const float4*)(side + base + i);
    float4 e = *(const float4*)(ego  + base + i);
    float4 p = make_float4(s.x * e.x, s.y * e.y, s.z * e.z, s.w * e.w);
    *(float4*)(&A1[r * ASTRIDE + c]) = s;
    *(float4*)(&A2[r * ASTRIDE + c]) = p;
  }
  if (tid < 32) rowss[tid] = 0.0f;
  __syncthreads();

  const int wave = tid >> 5;
  const int lane = tid & 31;
  const int m16  = lane & 15;
  const int hi   = lane >> 4;        // half-wave select
  const int koff = hi << 1;          // K offset 0 / 2 per fp32 A/B fragment layout
  const int n    = wave * 16 + m16;  // output column

  float bias = bgc[n] + bbi[n];
  v8f acc[2];
#pragma unroll
  for (int t = 0; t < 2; ++t)
#pragma unroll
    for (int v = 0; v < 8; ++v) acc[t][v] = bias;

  const float* a1p0 = &A1[m16 * ASTRIDE + koff];
  const float* a1p1 = &A1[(16 + m16) * ASTRIDE + koff];
  const float* a2p0 = &A2[m16 * ASTRIDE + koff];
  const float* a2p1 = &A2[(16 + m16) * ASTRIDE + koff];

  for (int kk = 0; kk < DIN; kk += 4) {
    int kr = kk + koff;
    v2f b1, b2, a;
    b1[0] = Wgc[kr * DIN + n];  b1[1] = Wgc[(kr + 1) * DIN + n];
    b2[0] = Wbi[kr * DIN + n];  b2[1] = Wbi[(kr + 1) * DIN + n];
    a[0] = a1p0[kk];  a[1] = a1p0[kk + 1];
    acc[0] = __builtin_amdgcn_wmma_f32_16x16x4_f32(false, a, false, b1, (short)0, acc[0], false, false);
    a[0] = a1p1[kk];  a[1] = a1p1[kk + 1];
    acc[1] = __builtin_amdgcn_wmma_f32_16x16x4_f32(false, a, false, b1, (short)0, acc[1], false, false);
    a[0] = a2p0[kk];  a[1] = a2p0[kk + 1];
    acc[0] = __builtin_amdgcn_wmma_f32_16x16x4_f32(false, a, false, b2, (short)0, acc[0], false, false);
    a[0] = a2p1[kk];  a[1] = a2p1[kk + 1];
    acc[1] = __builtin_amdgcn_wmma_f32_16x16x4_f32(false, a, false, b2, (short)0, acc[1], false, false);
  }

#pragma unroll
  for (int t = 0; t < 2; ++t) {
#pragma unroll
    for (int v = 0; v < 8; ++v) {
      float x = acc[t][v];
      acc[t][v] = (x > 0.0f) ? x : NEG_SLOPE * x;
    }
    // per-row sum of squares: reduce across the 16 lanes of each half-wave
#pragma unroll
    for (int v = 0; v < 8; ++v) {
      float x = acc[t][v] * acc[t][v];
      x += __shfl_xor(x, 1, 16);
      x += __shfl_xor(x, 2, 16);
      x += __shfl_xor(x, 4, 16);
      x += __shfl_xor(x, 8, 16);
      if (m16 == 0) atomicAdd(&rowss[t * 16 + v + hi * 8], x);
    }
  }
  __syncthreads();

#pragma unroll
  for (int t = 0; t < 2; ++t)
#pragma unroll
    for (int v = 0; v < 8; ++v) {
      int  M = t * 16 + v + hi * 8;
      float x = acc[t][v];
      long o = base + (long)M * DIN + n;
      if (out_raw) out_raw[o] = x;
      out_norm[o] = x / fmaxf(sqrtf(rowss[M]), 1e-12f);
    }
}

// Gather rows of [E0 | N1 | N2] for users / pos_items / neg_items -> (3*4096, 384)
__global__ void gather_kernel(const int* __restrict__ users,
                              const int* __restrict__ pos,
                              const int* __restrict__ neg,
                              const float* __restrict__ E0,
                              const float* __restrict__ N1,
                              const float* __restrict__ N2,
                              float* __restrict__ out) {
  int b = blockIdx.x;          // 0..3*4096-1
  int t = threadIdx.x;         // 0..127
  int which = b >> 12;         // 0=user, 1=pos, 2=neg
  int i = b & 4095;
  long row;
  if (which == 0)      row = users[i];
  else if (which == 1) row = (long)N_USER + pos[i];
  else                 row = (long)N_USER + neg[i];
  float* o = out + (long)b * 384;
  o[t]       = E0[row * DIN + t];
  o[t + 128] = N1[row * DIN + t];
  o[t + 256] = N2[row * DIN + t];
}

extern "C" void kernel_launch(void* const* d_in, const int* in_sizes, int n_in,
                              void* d_out, int out_size, void* d_ws, size_t ws_size,
                              hipStream_t stream) {
  (void)in_sizes; (void)n_in; (void)out_size; (void)ws_size;
  const int*   users    = (const int*)d_in[0];
  const int*   pos      = (const int*)d_in[1];
  const int*   neg      = (const int*)d_in[2];
  const float* user_emb = (const float*)d_in[3];
  const float* item1    = (const float*)d_in[4];
  const float* item2    = (const float*)d_in[5];
  const float* feat     = (const float*)d_in[6];
  const float* w        = (const float*)d_in[7];
  const float* Wgc      = (const float*)d_in[8];
  const float* bgc      = (const float*)d_in[9];
  const float* Wbi      = (const float*)d_in[10];
  const float* bbi      = (const float*)d_in[11];
  const int*   rows     = (const int*)d_in[12];
  const int*   cols     = (const int*)d_in[13];
  const float* vals     = (const float*)d_in[14];
  float* out = (float*)d_out;
  float* ws  = (float*)d_ws;

  const long NE = (long)N_TOT * DIN;      // 12.8M floats per matrix
  float* E0 = ws;                          // ego0 (all_emb[0])
  float* S  = ws + NE;                     // side (reused per layer)
  float* E1 = ws + 2 * NE;                 // layer-0 unnormalized ego
  float* N1 = ws + 3 * NE;                 // layer-0 normalized (all_emb[1])
  float* N2 = ws + 4 * NE;                 // layer-1 normalized (all_emb[2])
  float* CB = ws + 5 * NE;                 // comb vals (NNZ floats)

  build_ego_kernel<<<(int)(NE / 4 / 256), 256, 0, stream>>>(user_emb, item1, item2, feat, E0);
  comb_kernel<<<NNZ_C / 256, 256, 0, stream>>>(vals, w, CB);

  // layer 0
  hipMemsetAsync(S, 0, NE * sizeof(float), stream);
  spmm_kernel<<<NNZ_C / 8, 256, 0, stream>>>(CB, rows, cols, E0, S);
  gemm_layer_kernel<<<N_TOT / 32, 256, 0, stream>>>(S, E0, Wgc, bgc, Wbi, bbi, E1, N1);

  // layer 1
  hipMemsetAsync(S, 0, NE * sizeof(float), stream);
  spmm_kernel<<<NNZ_C / 8, 256, 0, stream>>>(CB, rows, cols, E1, S);
  gemm_layer_kernel<<<N_TOT / 32, 256, 0, stream>>>(S, E1, Wgc + DIN * DIN, bgc + DIN,
                                                    Wbi + DIN * DIN, bbi + DIN,
                                                    nullptr, N2);

  gather_kernel<<<3 * BATCH, 128, 0, stream>>>(users, pos, neg, E0, N1, N2, out);
}